// HungarianMatcher_37967510896919
// MI455X (gfx1250) — compile-verified
//
#include <hip/hip_runtime.h>
#include <hip/hip_bf16.h>
#include <cmath>

typedef __attribute__((ext_vector_type(2))) float v2f;
typedef __attribute__((ext_vector_type(8))) float v8f;

#define NQ 100
#define JD 51
#define JD4 13     // padded to 52 floats = 13 float4
#define NTILE 7    // ceil(100/16)

__global__ __launch_bounds__(256) void matcher_blocks_kernel(
    const float* __restrict__ pred_conf,   // (16,100,1)
    const float* __restrict__ pred_joint,  // (16,100,51)
    const float* __restrict__ pred_action, // (16,100,8)
    const float* __restrict__ pred_ident,  // (16,100,10)
    const float* __restrict__ tar_conf,    // (16,100)
    const float* __restrict__ tar_joint,   // (16,100,51)
    float* __restrict__ out)               // (16,100,100)
{
    __shared__ float4 sPJ4[NQ * JD4];  // pred_joint rows, padded to 52 floats
    __shared__ float4 sTJ4[NQ * JD4];  // tar_joint rows, padded to 52 floats
    __shared__ float  sDlp[NQ];        // log p - log(1-p) == logit == pred_conf
    __shared__ float  sL1mp[NQ];       // log(1-p)
    __shared__ float  sRow[NQ];        // ce_action + ce_identity (per pred row)
    __shared__ float  sNegT[NQ];       // -tar_conf

    const int b   = blockIdx.y;
    const int tid = threadIdx.x;

    // Stage joint tensors into LDS with row padding 51 -> 52 (pad value 0
    // contributes |0-0| = 0 to the L1 sum).
    {
        float* dPJ = reinterpret_cast<float*>(sPJ4);
        float* dTJ = reinterpret_cast<float*>(sTJ4);
        const float* pj = pred_joint + b * NQ * JD;
        const float* tj = tar_joint  + b * NQ * JD;
        for (int idx = tid; idx < NQ * (4 * JD4); idx += 256) {
            int row = idx / (4 * JD4);
            int col = idx - row * (4 * JD4);
            float vp = 0.0f, vt = 0.0f;
            if (col < JD) {
                vp = pj[row * JD + col];
                vt = tj[row * JD + col];
            }
            dPJ[idx] = vp;
            dTJ[idx] = vt;
        }
    }

    // Per-row / per-col scalar terms (100 threads).
    if (tid < NQ) {
        float x = pred_conf[b * NQ + tid];
        sDlp[tid] = x;  // log(p) - log(1-p) = x exactly
        // log(1 - sigmoid(x)) = -log(1+exp(x)), numerically stable split:
        float l1mp = (x > 0.0f) ? (-x - log1pf(__expf(-x)))
                                : (-log1pf(__expf(x)));
        sL1mp[tid] = l1mp;

        // ce = logsumexp(logits) - logits[0]  (reference takes column 0)
        const float* pa = pred_action + (b * NQ + tid) * 8;
        float m = pa[0];
        #pragma unroll
        for (int k = 1; k < 8; ++k) m = fmaxf(m, pa[k]);
        float s = 0.0f;
        #pragma unroll
        for (int k = 0; k < 8; ++k) s += __expf(pa[k] - m);
        float ce = (m + __logf(s)) - pa[0];

        const float* pi = pred_ident + (b * NQ + tid) * 10;
        float m2 = pi[0];
        #pragma unroll
        for (int k = 1; k < 10; ++k) m2 = fmaxf(m2, pi[k]);
        float s2 = 0.0f;
        #pragma unroll
        for (int k = 0; k < 10; ++k) s2 += __expf(pi[k] - m2);
        ce += (m2 + __logf(s2)) - pi[0];

        sRow[tid]  = ce;
        sNegT[tid] = -tar_conf[b * NQ + tid];
    }
    __syncthreads();

    // One wave per 16x16 output tile; tile choice is wave-uniform so EXEC
    // is all-ones for the WMMA.
    const int wave = tid >> 5;
    const int lane = tid & 31;
    const int tile = blockIdx.x * 8 + wave;
    if (tile >= NTILE * NTILE) return;
    const int ti  = tile / NTILE;
    const int tjx = tile % NTILE;
    const int i0 = ti * 16;
    const int j0 = tjx * 16;

    // A (16x4 f32): lanes 0-15 hold {K=0,K=1} for row M=lane; lanes 16-31 hold
    // {K=2,K=3} which we zero (so B's upper-lane contents are don't-care).
    // B (4x16 f32): lane holds column N=lane&15; lanes 0-15 carry K=0,1.
    v2f a;  a.x = 0.0f;  a.y = 0.0f;
    v2f bb; bb.x = 0.0f; bb.y = 0.0f;
    const int mlo = lane & 15;
    if (lane < 16) {
        int i = i0 + mlo;
        if (i < NQ) { a.x = sDlp[i]; a.y = sL1mp[i]; }
        int j = j0 + mlo;
        if (j < NQ) { bb.x = sNegT[j]; bb.y = -1.0f; }
    }

    // D[M,N] = x_M*(-t_N) + log(1-p_M)*(-1) = -(t*log p + (1-t)*log(1-p))
    v8f c = {};
    c = __builtin_amdgcn_wmma_f32_16x16x4_f32(
            /*neg_a=*/false, a, /*neg_b=*/false, bb,
            /*c_mod=*/(short)0, c, /*reuse_a=*/false, /*reuse_b=*/false);

    // C/D 16x16 f32 layout: lane<16 -> (M=r, N=lane); lane>=16 -> (M=r+8, N=lane-16)
    const int j     = j0 + (lane & 15);
    const int ibase = i0 + ((lane >> 4) << 3);
    float* outb = out + b * NQ * NQ;
    if (j < NQ) {
        // Hoist this lane's target row into registers (13 x float4).
        const float4* tjrow = &sTJ4[j * JD4];
        float4 tr[JD4];
        #pragma unroll
        for (int kk = 0; kk < JD4; ++kk) tr[kk] = tjrow[kk];

        #pragma unroll
        for (int r = 0; r < 8; ++r) {
            int i = ibase + r;
            if (i < NQ) {
                const float4* pjrow = &sPJ4[i * JD4];
                float l1 = 0.0f;
                #pragma unroll
                for (int kk = 0; kk < JD4; ++kk) {
                    float4 pv = pjrow[kk];
                    l1 += fabsf(pv.x - tr[kk].x);
                    l1 += fabsf(pv.y - tr[kk].y);
                    l1 += fabsf(pv.z - tr[kk].z);
                    l1 += fabsf(pv.w - tr[kk].w);
                }
                outb[i * NQ + j] = c[r] + l1 + sRow[i];
            }
        }
    }
}

extern "C" void kernel_launch(void* const* d_in, const int* in_sizes, int n_in,
                              void* d_out, int out_size, void* d_ws, size_t ws_size,
                              hipStream_t stream) {
    (void)in_sizes; (void)n_in; (void)d_ws; (void)ws_size; (void)out_size;
    const float* pred_conf   = (const float*)d_in[0];  // (16,100,1)
    const float* pred_joint  = (const float*)d_in[1];  // (16,100,51)
    const float* pred_action = (const float*)d_in[2];  // (16,100,8)
    const float* pred_ident  = (const float*)d_in[3];  // (16,100,10)
    const float* tar_conf    = (const float*)d_in[4];  // (16,100)
    const float* tar_joint   = (const float*)d_in[5];  // (16,100,51)
    // d_in[6] (tar_action) and d_in[7] (tar_identity) are dead in the reference.
    float* out = (float*)d_out;                        // (16,100,100)

    dim3 grid(NTILE, 16);   // 7 tile-groups (8 waves each cover 49 tiles) x 16 batches
    dim3 block(256);
    matcher_blocks_kernel<<<grid, block, 0, stream>>>(
        pred_conf, pred_joint, pred_action, pred_ident, tar_conf, tar_joint, out);
}